// DiscriminativeLoss_10411000725705
// MI455X (gfx1250) — compile-verified
//
#include <hip/hip_runtime.h>

typedef float v2f __attribute__((ext_vector_type(2)));
typedef float v4f __attribute__((ext_vector_type(4)));
typedef float v8f __attribute__((ext_vector_type(8)));
typedef int   v4i __attribute__((ext_vector_type(4)));

#define NB        4
#define ECH       16
#define NPIX      (1024 * 1024)
#define KLAB      16            // padded label slots (real labels 0..9)

// ---- workspace layout (floats) ----
// sums    [NB][16][16]  (b, label, channel)   @ 0      (1024)
// counts  [NB][16]                            @ 1024   (64)
// centers [NB][16][16]  (b, label, channel)   @ 1088   (1024)
// hsum    [NB][16]                            @ 2112   (64)
#define WS_FLOATS 2176

// --------------------------------------------------------------------------
__global__ void zero_ws_kernel(float* __restrict__ ws, int n) {
    int i = blockIdx.x * blockDim.x + threadIdx.x;
    if (i < n) ws[i] = 0.0f;
}

// --------------------------------------------------------------------------
// Pass 1: segmented sums via one-hot WMMA (f32 16x16x4) + LDS count histogram.
// Grid: 32 blocks per image * NB images, 256 threads (8 waves) per block.
// Each wave owns 4096 contiguous pixels of one image.
#define PIX_PER_WAVE   4096
#define WAVES_PER_BLK  8
#define BLKS_PER_IMG   (NPIX / (PIX_PER_WAVE * WAVES_PER_BLK))   // 32

__global__ void seg_sums_kernel(const float* __restrict__ emb,
                                const int*  __restrict__ lab,
                                float* __restrict__ sums,     // [NB][16][16]
                                float* __restrict__ counts) { // [NB][16]
    __shared__ unsigned scnt[KLAB];

    const int lane  = threadIdx.x & 31;
    const int wave  = threadIdx.x >> 5;
    const int b     = blockIdx.x / BLKS_PER_IMG;
    const int binb  = blockIdx.x % BLKS_PER_IMG;
    const int wIdx  = binb * WAVES_PER_BLK + wave;
    const int p0    = wIdx * PIX_PER_WAVE;

    if (threadIdx.x < KLAB) scnt[threadIdx.x] = 0u;
    __syncthreads();

    const float* embB = emb + (size_t)b * ECH * NPIX;
    const int*   labB = lab + (size_t)b * NPIX;

    const int half = lane >> 4;     // 0: lanes 0-15, 1: lanes 16-31
    const int mrow = lane & 15;     // A: channel row (M); B/C: label column (N)
    const int kOff = half * 2;      // A/B K-pair held by this half-wave

    v8f acc = {0.f, 0.f, 0.f, 0.f, 0.f, 0.f, 0.f, 0.f};

    for (int p = p0; p < p0 + PIX_PER_WAVE; p += 32) {
        int lv = labB[p + lane] & 15;          // lane's pixel label
        atomicAdd(&scnt[lv], 1u);              // counts histogram (ds_add)

        #pragma unroll
        for (int g = 0; g < 8; ++g) {          // 8 groups of 4 pixels
            const int kbase = 4 * g + kOff;

            // labels of pixels (kbase, kbase+1): direct broadcast load,
            // hits the label cacheline already in L0 (no cross-lane ds ops)
            const int2 lw = *reinterpret_cast<const int2*>(labB + p + kbase);

            // A tile: emb[channel=mrow][p + kbase .. +1]  (b64 per lane)
            const float2 av =
                *reinterpret_cast<const float2*>(embB + (size_t)mrow * NPIX + p + kbase);
            v2f a; a[0] = av.x; a[1] = av.y;

            // B tile: one-hot(label == column)
            v2f bm;
            bm[0] = ((lw.x & 15) == mrow) ? 1.0f : 0.0f;
            bm[1] = ((lw.y & 15) == mrow) ? 1.0f : 0.0f;

            // C[channel][label] += A x B  (exact f32 accumulation)
            acc = __builtin_amdgcn_wmma_f32_16x16x4_f32(
                false, a, false, bm, (short)0, acc, false, false);
        }
    }

    // C layout: VGPR r -> M = r + 8*half (channel), N = mrow (label).
    #pragma unroll
    for (int r = 0; r < 8; ++r) {
        const int ch = r + 8 * half;
        atomicAdd(&sums[((size_t)b * 16 + mrow) * 16 + ch], acc[r]);
    }

    __syncthreads();
    if (threadIdx.x < KLAB)
        atomicAdd(&counts[b * 16 + threadIdx.x], (float)scnt[threadIdx.x]);
}

// --------------------------------------------------------------------------
__global__ void centers_kernel(const float* __restrict__ sums,
                               const float* __restrict__ counts,
                               float* __restrict__ centers) {
    int i = blockIdx.x * blockDim.x + threadIdx.x;     // over NB*16*16
    if (i < NB * 16 * 16) {
        float c = counts[i >> 4];
        centers[i] = sums[i] / (c > 0.0f ? c : 1.0f);
    }
}

// --------------------------------------------------------------------------
// Pass 2: per-pixel variance hinge, segment-summed per label.
// 4 consecutive pixels per lane -> b128 streaming loads, non-temporal (no reuse).
#define PIX_PER_BLK2  8192
#define BLKS2_PER_IMG (NPIX / PIX_PER_BLK2)            // 128
#define CPAD          17                               // LDS bank-conflict padding

__global__ void var_kernel(const float* __restrict__ emb,
                           const int*  __restrict__ lab,
                           const float* __restrict__ centers,
                           float* __restrict__ hsum) {  // [NB][16]
    __shared__ float cLds[KLAB * CPAD];
    __shared__ float hLds[KLAB];

    const int b  = blockIdx.x / BLKS2_PER_IMG;
    const int p0 = (blockIdx.x % BLKS2_PER_IMG) * PIX_PER_BLK2;

    {   // centers -> padded LDS tile
        const int k = threadIdx.x >> 4, e = threadIdx.x & 15;   // blockDim == 256
        cLds[k * CPAD + e] = centers[(size_t)b * 256 + threadIdx.x];
        if (threadIdx.x < KLAB) hLds[threadIdx.x] = 0.0f;
    }
    __syncthreads();

    const float* embB = emb + (size_t)b * ECH * NPIX;
    const int*   labB = lab + (size_t)b * NPIX;

    for (int p = p0 + threadIdx.x * 4; p < p0 + PIX_PER_BLK2; p += 256 * 4) {
        const v4i lv4 = __builtin_nontemporal_load(
            reinterpret_cast<const v4i*>(labB + p));
        const int c0 = (lv4[0] & 15) * CPAD;
        const int c1 = (lv4[1] & 15) * CPAD;
        const int c2 = (lv4[2] & 15) * CPAD;
        const int c3 = (lv4[3] & 15) * CPAD;

        float sq0 = 0.f, sq1 = 0.f, sq2 = 0.f, sq3 = 0.f;
        #pragma unroll
        for (int e = 0; e < ECH; ++e) {
            const v4f ev = __builtin_nontemporal_load(
                reinterpret_cast<const v4f*>(embB + (size_t)e * NPIX + p));
            float d0 = ev[0] - cLds[c0 + e]; sq0 = fmaf(d0, d0, sq0);
            float d1 = ev[1] - cLds[c1 + e]; sq1 = fmaf(d1, d1, sq1);
            float d2 = ev[2] - cLds[c2 + e]; sq2 = fmaf(d2, d2, sq2);
            float d3 = ev[3] - cLds[c3 + e]; sq3 = fmaf(d3, d3, sq3);
        }
        float t0 = fmaxf(sqrtf(fmaxf(sq0, 1e-30f)) - 0.5f, 0.f);
        float t1 = fmaxf(sqrtf(fmaxf(sq1, 1e-30f)) - 0.5f, 0.f);
        float t2 = fmaxf(sqrtf(fmaxf(sq2, 1e-30f)) - 0.5f, 0.f);
        float t3 = fmaxf(sqrtf(fmaxf(sq3, 1e-30f)) - 0.5f, 0.f);
        atomicAdd(&hLds[lv4[0] & 15], t0 * t0);
        atomicAdd(&hLds[lv4[1] & 15], t1 * t1);
        atomicAdd(&hLds[lv4[2] & 15], t2 * t2);
        atomicAdd(&hLds[lv4[3] & 15], t3 * t3);
    }

    __syncthreads();
    if (threadIdx.x < KLAB)
        atomicAdd(&hsum[b * 16 + threadIdx.x], hLds[threadIdx.x]);
}

// --------------------------------------------------------------------------
__global__ void final_kernel(const float* __restrict__ counts,
                             const float* __restrict__ hsum,
                             const float* __restrict__ centers,
                             float* __restrict__ out) {
    __shared__ float sVar[NB], sDst[NB], sReg[NB], sNi[NB], sNp[NB];
    const int t = threadIdx.x;
    if (t < NB) { sVar[t] = 0; sDst[t] = 0; sReg[t] = 0; sNi[t] = 0; sNp[t] = 0; }
    __syncthreads();

    // per-(b, label): variance + regularization terms
    if (t < NB * 16) {
        const int b = t >> 4, k = t & 15;
        const float cN = counts[b * 16 + k];
        if (cN > 0.0f && k > 0 && k < 10) {
            atomicAdd(&sNi[b], 1.0f);
            atomicAdd(&sVar[b], hsum[b * 16 + k] / cN);
            float cn2 = 0.0f;
            for (int e = 0; e < ECH; ++e) {
                float c = centers[((size_t)b * 16 + k) * 16 + e];
                cn2 = fmaf(c, c, cn2);
            }
            atomicAdd(&sReg[b], sqrtf(cn2));
        }
    }

    // per-(b, pair): distance hinge over unordered label pairs i < j in 0..9
    if (t < NB * 45) {
        const int b = t / 45;
        int rem = t % 45, i = 0;
        while (rem >= 9 - i) { rem -= 9 - i; ++i; }
        const int j = i + 1 + rem;
        const bool pi = (counts[b * 16 + i] > 0.0f) && (i > 0);
        const bool pj = (counts[b * 16 + j] > 0.0f);
        if (pi && pj) {
            float sq = 0.0f;
            for (int e = 0; e < ECH; ++e) {
                float d = centers[((size_t)b * 16 + i) * 16 + e]
                        - centers[((size_t)b * 16 + j) * 16 + e];
                sq = fmaf(d, d, sq);
            }
            float h = fmaxf(3.0f - sqrtf(sq), 0.0f);   // 2*DELTA_DST = 3
            atomicAdd(&sDst[b], h * h);
            atomicAdd(&sNp[b], 1.0f);
        }
    }

    __syncthreads();
    if (t == 0) {
        float vb = 0, lv = 0, ld = 0, lr = 0;
        for (int b = 0; b < NB; ++b) {
            const float ni = sNi[b];
            const float valid = ni > 0.0f ? 1.0f : 0.0f;
            const float di = fmaxf(ni, 1.0f);
            const float dp = fmaxf(sNp[b], 1.0f);
            lv += valid * (sVar[b] / di);
            ld += valid * (sDst[b] / dp);
            lr += valid * (sReg[b] / di);
            vb += valid;
        }
        vb = fmaxf(vb, 1.0f);
        lv /= vb; ld /= vb; lr /= vb;
        out[0] = lv + ld + 0.001f * lr;   // A_W=1, B_W=1, R_W=0.001
        out[1] = lv;
        out[2] = ld;
        out[3] = lr;
    }
}

// --------------------------------------------------------------------------
extern "C" void kernel_launch(void* const* d_in, const int* in_sizes, int n_in,
                              void* d_out, int out_size, void* d_ws, size_t ws_size,
                              hipStream_t stream) {
    const float* emb = (const float*)d_in[0];
    const int*   lab = (const int*)d_in[1];
    float* out = (float*)d_out;
    float* ws  = (float*)d_ws;

    float* sums    = ws;          // 1024
    float* counts  = ws + 1024;   // 64
    float* centers = ws + 1088;   // 1024
    float* hsum    = ws + 2112;   // 64

    zero_ws_kernel<<<(WS_FLOATS + 255) / 256, 256, 0, stream>>>(ws, WS_FLOATS);

    seg_sums_kernel<<<NB * BLKS_PER_IMG, 256, 0, stream>>>(emb, lab, sums, counts);

    centers_kernel<<<(NB * 256 + 255) / 256, 256, 0, stream>>>(sums, counts, centers);

    var_kernel<<<NB * BLKS2_PER_IMG, 256, 0, stream>>>(emb, lab, centers, hsum);

    final_kernel<<<1, 256, 0, stream>>>(counts, hsum, centers, out);
}